// ProbEstimation_19232863552211
// MI455X (gfx1250) — compile-verified
//
#include <hip/hip_runtime.h>

// Problem constants (from reference): inputs [16, 1024, 1024] f32, top-5, BW=2
#define BDIM   16
#define TDIM   1024
#define DDIM   1024
#define NTOP   5
// 1 / (BW * sqrt(2*pi)) with BW = 2
#define NORMC  0.19947114020071635f

typedef __attribute__((ext_vector_type(2))) float v2f;
typedef __attribute__((ext_vector_type(8))) float v8f;

// ---------------------------------------------------------------------------
// Kernel 0: zero the histogram workspace (d_ws is poisoned with 0xAA).
// ---------------------------------------------------------------------------
__global__ void pe_zero_hist(float* __restrict__ H) {
    int i = blockIdx.x * blockDim.x + threadIdx.x;
    if (i < BDIM * DDIM) H[i] = 0.0f;
}

// ---------------------------------------------------------------------------
// Kernel 1: one wave32 per (b2, t) row. Stream the 1024-float row once
// (float4 coalesced), find top-5 indices (ties -> smaller index, matching
// jax.lax.top_k stability), and bump the histogram row hb = t >> 6 with
// global_atomic_add_f32. Counts are small integers -> exact in f32.
//
// Lane-local layout: slot s = 4*i + c holds element g = 128*i + c + 4*lane.
// Within a lane, ascending s <=> ascending g, so a strict '>' scan picks the
// smallest index among lane-local ties. A 32-bit kill mask removes winners
// (avoids dynamic register-array indexing).
// ---------------------------------------------------------------------------
__global__ __launch_bounds__(256) void pe_topk_hist(const float* __restrict__ in,
                                                    float* __restrict__ H) {
    const int lane = threadIdx.x & 31;
    const int wid  = threadIdx.x >> 5;
    const int row  = blockIdx.x * 8 + wid;      // (b2, t) row in [0, 16384)
    const int t    = row & (TDIM - 1);
    const int hb   = t >> 6;                    // destination histogram row

    const float4* rp = (const float4*)(in + (size_t)row * DDIM);

    float v[32];
#pragma unroll
    for (int i = 0; i < 8; ++i) {
        float4 q = rp[i * 32 + lane];           // 512B contiguous per instr
        v[i * 4 + 0] = q.x;
        v[i * 4 + 1] = q.y;
        v[i * 4 + 2] = q.z;
        v[i * 4 + 3] = q.w;
    }

    const float NEG_INF = -__builtin_huge_valf();
    unsigned killed = 0u;

#pragma unroll
    for (int k = 0; k < NTOP; ++k) {
        // lane-local argmax over live slots
        float best = NEG_INF;
        int   bs   = 0;
#pragma unroll
        for (int s = 0; s < 32; ++s) {
            float val = ((killed >> s) & 1u) ? NEG_INF : v[s];
            if (val > best) { best = val; bs = s; }
        }
        int gi = ((bs >> 2) << 7) + (bs & 3) + (lane << 2);  // global column

        // wave-wide argmax butterfly; ties -> smaller global index
#pragma unroll
        for (int off = 16; off >= 1; off >>= 1) {
            float ov = __shfl_xor(best, off, 32);
            int   og = __shfl_xor(gi,   off, 32);
            if (ov > best || (ov == best && og < gi)) { best = ov; gi = og; }
        }
        // all lanes now agree on the winner gi
        if (lane == 0) atomicAdd(&H[hb * DDIM + gi], 1.0f);

        // owning lane kills the winning slot
        if (((gi >> 2) & 31) == lane) {
            int s = ((gi >> 7) << 2) | (gi & 3);
            killed |= (1u << s);
        }
    }
}

// ---------------------------------------------------------------------------
// Kernel 2: out[16,1024] = H[16,1024] x K[1024,1024] via V_WMMA_F32_16X16X4_F32.
// K[v,d] = NORMC * exp(-((d - v)^2)/8) is generated on the fly (2 v_exp_f32
// per lane per WMMA). One wave per 16-column output tile (64 waves), K-loop of
// 256 WMMA steps, f32 accumulate -- bit-equivalent math to the reference
// (summation reordering only).
//
// Fragment layouts (wave32, CDNA5 ISA 7.12.2):
//   A 16x4 f32 : lanes 0-15 -> M=lane,    v0=K0 v1=K1 ; lanes 16-31 -> K2,K3
//   B 4x16 f32 : lanes 0-15 -> N=lane,    v0=K0 v1=K1 ; lanes 16-31 -> K2,K3
//   C/D 16x16  : vgpr r, lanes 0-15 -> M=r, N=lane ; lanes 16-31 -> M=r+8
// ---------------------------------------------------------------------------
__global__ __launch_bounds__(32) void pe_gemm_wmma(const float* __restrict__ H,
                                                   float* __restrict__ out) {
    const int lane = threadIdx.x & 31;
    const int n0   = blockIdx.x << 4;     // output column tile base
    const int hi   = lane >> 4;           // 0 / 1 half of the wave
    const int lo   = lane & 15;           // m (for A) or n (for B/C/D)
    const int kb   = hi << 1;             // K offset handled by this half

    v8f c = {};
    const float dval = (float)(n0 + lo);

    for (int k0 = 0; k0 < DDIM; k0 += 4) {
        v2f a;
        a.x = H[lo * DDIM + (k0 + kb)];
        a.y = H[lo * DDIM + (k0 + kb + 1)];

        float d0 = dval - (float)(k0 + kb);
        float d1 = d0 - 1.0f;
        v2f b;
        b.x = NORMC * __expf(-0.125f * d0 * d0);
        b.y = NORMC * __expf(-0.125f * d1 * d1);

        c = __builtin_amdgcn_wmma_f32_16x16x4_f32(
                /*neg_a=*/false, a, /*neg_b=*/false, b,
                /*c_mod=*/(short)0, c, /*reuse_a=*/false, /*reuse_b=*/false);
    }

#pragma unroll
    for (int r = 0; r < 8; ++r) {
        out[(hi * 8 + r) * DDIM + n0 + lo] = c[r];
    }
}

// ---------------------------------------------------------------------------
// kernel_launch: zero H (64KB of d_ws) -> top-k histogram -> WMMA GEMM.
// ---------------------------------------------------------------------------
extern "C" void kernel_launch(void* const* d_in, const int* in_sizes, int n_in,
                              void* d_out, int out_size, void* d_ws, size_t ws_size,
                              hipStream_t stream) {
    (void)in_sizes; (void)n_in; (void)out_size; (void)ws_size;

    const float* in  = (const float*)d_in[0];       // [16, 1024, 1024] f32
    float*       out = (float*)d_out;               // [16, 1024] f32
    float*       H   = (float*)d_ws;                // [16, 1024] f32 histogram

    pe_zero_hist<<<(BDIM * DDIM + 255) / 256, 256, 0, stream>>>(H);
    pe_topk_hist<<<(BDIM * TDIM) / 8, 256, 0, stream>>>(in, H);
    pe_gemm_wmma<<<DDIM / 16, 32, 0, stream>>>(H, out);
}